// StructuredKMeans_57810259804652
// MI455X (gfx1250) — compile-verified
//
#include <hip/hip_runtime.h>

typedef __attribute__((ext_vector_type(2))) float v2f;
typedef __attribute__((ext_vector_type(4))) float v4f;
typedef __attribute__((ext_vector_type(8))) float v8f;
typedef __attribute__((ext_vector_type(4))) int   v4i;

#define B_ROWS 131072
#define C_CENT 1024
#define D_DIM  512
#define MT     32          // rows of x per block
#define KC     16          // K chunk staged in LDS per iteration
#define NITER  (D_DIM / KC)
#define AS_STRIDE 516      // 512 + 4 pad  (bank spread, keeps 16B alignment)
#define BS_STRIDE 20       // 16 + 4 pad   (conflict-free b64 fragment reads)
#define DS_STRIDE 1028     // 1024 + 4 pad (conflict-free dist stores)
#define BS_ELEMS  (C_CENT * BS_STRIDE)

#ifndef __has_builtin
#define __has_builtin(x) 0
#endif

#if __has_builtin(__builtin_amdgcn_global_load_async_to_lds_b128)
#define HAS_ASYNC_LDS 1
typedef __attribute__((address_space(1))) v4i g_v4i;   // global int4 (param 1)
typedef __attribute__((address_space(3))) v4i l_v4i;   // LDS int4    (param 2)
#else
#define HAS_ASYNC_LDS 0
#endif

#if HAS_ASYNC_LDS
#if __has_builtin(__builtin_amdgcn_s_wait_asynccnt)
#define WAIT_ASYNC() __builtin_amdgcn_s_wait_asynccnt(0)
#else
#define WAIT_ASYNC() asm volatile("s_wait_asynccnt 0x0" ::: "memory")
#endif
#else
#define WAIT_ASYNC() ((void)0)
#endif

struct SMem {
  union {
    struct {
      float As[MT * AS_STRIDE];        // 66,048 B : x tile, full K
      float Bs[2 * BS_ELEMS];          // 163,840 B : double-buffered centers chunk
    } ab;
    float dist[MT * DS_STRIDE];        // 131,584 B : distance overlay (post-GEMM)
  };
  float xpart[MT * 8];
  float xsq[MT];
};

__global__ void StructuredKMeans_csq_kernel(const float* __restrict__ centers,
                                            float* __restrict__ csq) {
  const int c = blockIdx.x;
  const int lane = threadIdx.x;
  const float* row = centers + (size_t)c * D_DIM;
  float s = 0.f;
  #pragma unroll
  for (int i = 0; i < D_DIM / 32; ++i) {
    float v = row[lane + 32 * i];
    s += v * v;
  }
  #pragma unroll
  for (int off = 16; off > 0; off >>= 1) s += __shfl_xor(s, off, 32);
  if (lane == 0) csq[c] = s;
}

// ---- stage one [1024 x KC] centers chunk into an LDS buffer ----
__device__ __forceinline__ void stage_Bs(const float* __restrict__ centers,
                                         float* __restrict__ bsBuf,
                                         int kc, int tid) {
#if HAS_ASYNC_LDS
  #pragma unroll
  for (int it = 0; it < (C_CENT * KC / 4) / 256; ++it) {   // 16 async b128 per thread
    int id = it * 256 + tid;
    int n  = id >> 2;
    int c4 = id & 3;
    __builtin_amdgcn_global_load_async_to_lds_b128(
        (g_v4i*)(centers + (size_t)n * D_DIM + kc + c4 * 4),
        (l_v4i*)(bsBuf + n * BS_STRIDE + c4 * 4), 0, 0);
  }
#else
  // pipelined fallback: 8 loads in independent regs, then 8 LDS stores
  #pragma unroll
  for (int g = 0; g < 2; ++g) {
    v4f t[8];
    #pragma unroll
    for (int u = 0; u < 8; ++u) {
      int id = (g * 8 + u) * 256 + tid;
      int n  = id >> 2;
      int c4 = id & 3;
      t[u] = *(const v4f*)(centers + (size_t)n * D_DIM + kc + c4 * 4);
    }
    #pragma unroll
    for (int u = 0; u < 8; ++u) {
      int id = (g * 8 + u) * 256 + tid;
      int n  = id >> 2;
      int c4 = id & 3;
      *(v4f*)(bsBuf + n * BS_STRIDE + c4 * 4) = t[u];
    }
  }
#endif
}

// ---- stage x tile [MT x 512] into LDS ----
__device__ __forceinline__ void stage_As(const float* __restrict__ x,
                                         float* __restrict__ asBuf, int tid) {
#if HAS_ASYNC_LDS
  #pragma unroll
  for (int it = 0; it < (MT * D_DIM / 4) / 256; ++it) {    // 16 async b128 per thread
    int id = it * 256 + tid;
    int r  = id >> 7;
    int c4 = id & 127;
    __builtin_amdgcn_global_load_async_to_lds_b128(
        (g_v4i*)(x + (size_t)r * D_DIM + c4 * 4),
        (l_v4i*)(asBuf + r * AS_STRIDE + c4 * 4), 0, 0);
  }
#else
  #pragma unroll
  for (int g = 0; g < 2; ++g) {
    v4f t[8];
    #pragma unroll
    for (int u = 0; u < 8; ++u) {
      int id = (g * 8 + u) * 256 + tid;
      int r  = id >> 7;
      int c4 = id & 127;
      t[u] = *(const v4f*)(x + (size_t)r * D_DIM + c4 * 4);
    }
    #pragma unroll
    for (int u = 0; u < 8; ++u) {
      int id = (g * 8 + u) * 256 + tid;
      int r  = id >> 7;
      int c4 = id & 127;
      *(v4f*)(asBuf + r * AS_STRIDE + c4 * 4) = t[u];
    }
  }
#endif
}

// ---- one K-chunk of WMMAs: batched B-fragment loads, then 16 WMMAs per k0 ----
__device__ __forceinline__ void compute_chunk(const float* __restrict__ As,
                                              const float* __restrict__ Bs,
                                              int kc, int l16, int half, int nbase,
                                              v8f acc[2][8]) {
  #pragma unroll
  for (int k0 = 0; k0 < KC; k0 += 4) {
    const int koff = k0 + 2 * half;          // lane-half selects K pair (ISA layout)
    v2f a0 = *(const v2f*)&As[(0  + l16) * AS_STRIDE + kc + koff];
    v2f a1 = *(const v2f*)&As[(16 + l16) * AS_STRIDE + kc + koff];
    v2f b[8];
    #pragma unroll
    for (int j = 0; j < 8; ++j)
      b[j] = *(const v2f*)&Bs[(nbase + j * 16 + l16) * BS_STRIDE + koff];
    #pragma unroll
    for (int j = 0; j < 8; ++j) {
      acc[0][j] = __builtin_amdgcn_wmma_f32_16x16x4_f32(
          false, a0, false, b[j], (short)0, acc[0][j], false, false);
      acc[1][j] = __builtin_amdgcn_wmma_f32_16x16x4_f32(
          false, a1, false, b[j], (short)0, acc[1][j], false, false);
    }
  }
}

__global__ __launch_bounds__(256)
void StructuredKMeans_fused_kernel(const float* __restrict__ x,
                                   const float* __restrict__ centers,
                                   const float* __restrict__ csq,
                                   float* __restrict__ dist_out,
                                   float* __restrict__ prob_out) {
  __shared__ SMem sm;
  const int tid  = threadIdx.x;
  const int lane = tid & 31;
  const int wave = tid >> 5;
  const int half = lane >> 4;
  const int l16  = lane & 15;
  const size_t rowbase = (size_t)blockIdx.x * MT;

  float* bufA = sm.ab.Bs;                 // double buffer pointers
  float* bufB = sm.ab.Bs + BS_ELEMS;

  // ---- prologue: stage x tile + first centers chunk ----
  stage_As(x + rowbase * D_DIM, sm.ab.As, tid);
  stage_Bs(centers, bufA, 0, tid);
  WAIT_ASYNC();
  __syncthreads();

  // ---- x row squared norms (8 threads per row -> reduce) ----
  {
    int r = tid >> 3, seg = tid & 7;
    const float* p = &sm.ab.As[r * AS_STRIDE + seg * 64];
    float s = 0.f;
    #pragma unroll
    for (int i = 0; i < 64; ++i) s += p[i] * p[i];
    sm.xpart[r * 8 + seg] = s;
  }
  __syncthreads();
  if (tid < MT) {
    float s = 0.f;
    #pragma unroll
    for (int i = 0; i < 8; ++i) s += sm.xpart[tid * 8 + i];
    sm.xsq[tid] = s;
  }

  // ---- per-wave column strip: preload ||c||^2 for this lane's columns ----
  const int nbase = wave * 128;
  float creg[8];
  #pragma unroll
  for (int j = 0; j < 8; ++j) creg[j] = csq[nbase + j * 16 + l16];

  const v8f vzero = {};
  v8f acc[2][8];
  #pragma unroll
  for (int mt = 0; mt < 2; ++mt)
    #pragma unroll
    for (int j = 0; j < 8; ++j) acc[mt][j] = vzero;

  // ---- K loop: prefetch next chunk async while computing current ----
  float* cur = bufA;
  float* nxt = bufB;
  for (int it = 0; it < NITER; ++it) {
    const int kc = it * KC;
    if (it + 1 < NITER) stage_Bs(centers, nxt, kc + KC, tid);
    compute_chunk(sm.ab.As, cur, kc, l16, half, nbase, acc);
    WAIT_ASYNC();            // own prefetch complete (LDS data written)
    __syncthreads();         // everyone done computing `cur`; `nxt` ready for all
    float* t = cur; cur = nxt; nxt = t;
  }

  // ---- epilogue 1: distances -> LDS overlay ----
  #pragma unroll
  for (int mt = 0; mt < 2; ++mt) {
    const int m0 = mt * 16;
    #pragma unroll
    for (int j = 0; j < 8; ++j) {
      const int nc = nbase + j * 16 + l16;
      #pragma unroll
      for (int i = 0; i < 8; ++i) {
        const int m = m0 + i + 8 * half;   // C/D layout: VGPR i = rows i / i+8 per half
        float d = sm.xsq[m] + creg[j] - 2.0f * acc[mt][j][i];
        d = fmaxf(d, 0.0f);
        sm.dist[m * DS_STRIDE + nc] = d;
      }
    }
  }
  __syncthreads();

  // ---- epilogue 2a: coalesced b128 store of distances ----
  {
    float* dptr = dist_out + rowbase * C_CENT;
    #pragma unroll
    for (int it = 0; it < (MT * C_CENT / 4) / 256; ++it) {   // 32 iters
      int id = it * 256 + tid;
      int r  = id >> 8;
      int c4 = id & 255;
      v4f v = *(const v4f*)(&sm.dist[r * DS_STRIDE + c4 * 4]);
      *(v4f*)(dptr + (size_t)r * C_CENT + c4 * 4) = v;
    }
  }

  // ---- epilogue 2b: row softmax(-d), 4 rows per wave ----
  {
    float* pptr = prob_out + rowbase * C_CENT;
    for (int rr = 0; rr < 4; ++rr) {
      const int r = wave * 4 + rr;
      const float* drow = &sm.dist[r * DS_STRIDE];
      float mn = 3.402823466e38f;
      #pragma unroll
      for (int i = 0; i < 32; ++i) mn = fminf(mn, drow[lane + 32 * i]);
      #pragma unroll
      for (int off = 16; off > 0; off >>= 1) mn = fminf(mn, __shfl_xor(mn, off, 32));
      float e[32];
      float s = 0.f;
      #pragma unroll
      for (int i = 0; i < 32; ++i) {
        float ev = __expf(mn - drow[lane + 32 * i]);   // exponents <= 0
        e[i] = ev;
        s += ev;
      }
      #pragma unroll
      for (int off = 16; off > 0; off >>= 1) s += __shfl_xor(s, off, 32);
      const float inv = 1.0f / s;
      #pragma unroll
      for (int i = 0; i < 32; ++i)
        pptr[(size_t)r * C_CENT + lane + 32 * i] = e[i] * inv;
    }
  }
}

extern "C" void kernel_launch(void* const* d_in, const int* in_sizes, int n_in,
                              void* d_out, int out_size, void* d_ws, size_t ws_size,
                              hipStream_t stream) {
  const float* x       = (const float*)d_in[0];
  const float* centers = (const float*)d_in[1];
  float* out      = (float*)d_out;
  float* dist_out = out;                                   // [B, C]
  float* prob_out = out + (size_t)B_ROWS * C_CENT;         // [B, C]
  float* csq      = (float*)d_ws;                          // [C]

  StructuredKMeans_csq_kernel<<<C_CENT, 32, 0, stream>>>(centers, csq);
  StructuredKMeans_fused_kernel<<<B_ROWS / MT, 256, 0, stream>>>(
      x, centers, csq, dist_out, prob_out);
}